// MatrixFunctionBlock_60146722013467
// MI455X (gfx1250) — compile-verified
//
#include <hip/hip_runtime.h>
#include <math.h>

// Problem constants (from reference):
#define B_   8
#define NG_  256
#define F_   8
#define P_   4
#define S_   128
#define NB_  8
#define FP_  32      // F*P
#define N_   2048    // B*NG
#define E_   32768
#define NMAT 256     // matrix dim n == NG
#define LDM  257     // padded LDS row stride -> bank = (i+j)%64, conflict-free

typedef float v2f __attribute__((ext_vector_type(2)));
typedef float v8f __attribute__((ext_vector_type(8)));

__device__ __forceinline__ float silu_(float x) { return x / (1.0f + __expf(-x)); }

// ---------------------------------------------------------------------------
// K1: node_s = node_feats @ W_scalar / sqrt(S) + b_scalar   (2048 x 128)
// ---------------------------------------------------------------------------
__global__ void k_node_s(const float* __restrict__ nf, const float* __restrict__ W,
                         const float* __restrict__ b, float* __restrict__ ns) {
  int row = blockIdx.x, t = threadIdx.x;
  const float* x = nf + (size_t)row * S_;
  float acc = 0.f;
  for (int k = 0; k < S_; ++k) acc += x[k] * W[k * S_ + t];
  ns[(size_t)row * S_ + t] = acc * 0.08838834764831845f + b[t];
}

// ---------------------------------------------------------------------------
// K2: per-edge MLPs + scatter-add into dense H [b][f][i][j]
// One 64-thread block per edge; hidden vectors live in LDS, weights stream
// from global (L2-resident broadcast).
// ---------------------------------------------------------------------------
__global__ void k_edge(const float* __restrict__ ef, const int* __restrict__ eidx,
                       const int* __restrict__ batch, const int* __restrict__ ptr,
                       const float* __restrict__ ns,
                       const float* __restrict__ We1, const float* __restrict__ We2,
                       const float* __restrict__ We3, const float* __restrict__ We4,
                       const float* __restrict__ Wm1, const float* __restrict__ Wm2,
                       const float* __restrict__ Wm3, float* __restrict__ H) {
  __shared__ float sa[128];
  __shared__ float sb[128];
  int e = blockIdx.x, t = threadIdx.x;
  int snd = eidx[e], rcv = eidx[E_ + e];
  if (t < NB_) sa[t] = ef[(size_t)e * NB_ + t];
  __syncthreads();
  // edge MLP: 8 -> 64 -> 64 -> 64 -> 128 (silu on all but last)
  float acc = 0.f;
  for (int k = 0; k < NB_; ++k) acc += sa[k] * We1[k * 64 + t];
  float h = silu_(acc * 0.35355339059327373f);          // /sqrt(8)
  __syncthreads(); sb[t] = h; __syncthreads();
  acc = 0.f; for (int k = 0; k < 64; ++k) acc += sb[k] * We2[k * 64 + t];
  h = silu_(acc * 0.125f);                              // /sqrt(64)
  __syncthreads(); sa[t] = h; __syncthreads();
  acc = 0.f; for (int k = 0; k < 64; ++k) acc += sa[k] * We3[k * 64 + t];
  h = silu_(acc * 0.125f);
  __syncthreads(); sb[t] = h; __syncthreads();
  float o0 = 0.f, o1 = 0.f;
  for (int k = 0; k < 64; ++k) { float v = sb[k]; o0 += v * We4[k * 128 + t]; o1 += v * We4[k * 128 + t + 64]; }
  o0 *= 0.125f; o1 *= 0.125f;                           // last layer, no silu
  // sym = node_s[snd] * node_s[rcv] * edge_w
  float p0 = ns[(size_t)snd * S_ + t]      * ns[(size_t)rcv * S_ + t]      * o0;
  float p1 = ns[(size_t)snd * S_ + t + 64] * ns[(size_t)rcv * S_ + t + 64] * o1;
  __syncthreads(); sa[t] = p0; sa[t + 64] = p1; __syncthreads();
  // matrix MLP: 128 -> 64 -> 64 -> 8
  acc = 0.f; for (int k = 0; k < 128; ++k) acc += sa[k] * Wm1[k * 64 + t];
  h = silu_(acc * 0.08838834764831845f);                // /sqrt(128)
  __syncthreads(); sb[t] = h; __syncthreads();
  acc = 0.f; for (int k = 0; k < 64; ++k) acc += sb[k] * Wm2[k * 64 + t];
  h = silu_(acc * 0.125f);
  __syncthreads(); sa[t] = h; __syncthreads();
  if (t < F_) {
    acc = 0.f; for (int k = 0; k < 64; ++k) acc += sa[k] * Wm3[k * F_ + t];
    acc *= 0.125f;
    int g = batch[snd];
    int li = snd - ptr[g], lj = rcv - ptr[g];
    atomicAdd(&H[(((size_t)g * F_ + t) * NMAT + li) * NMAT + lj], acc);
  }
}

// ---------------------------------------------------------------------------
// K3: degree[bf][i] = sum_j |H[bf][i][j]|
// ---------------------------------------------------------------------------
__global__ void k_deg(const float* __restrict__ H, float* __restrict__ deg) {
  __shared__ float red[NMAT];
  int row = blockIdx.x, t = threadIdx.x;
  red[t] = fabsf(H[(size_t)row * NMAT + t]);
  __syncthreads();
  for (int s = 128; s > 0; s >>= 1) { if (t < s) red[t] += red[t + s]; __syncthreads(); }
  if (t == 0) deg[row] = red[0];
}

// ---------------------------------------------------------------------------
// K4: per-channel (256 channels) — the hot kernel.
//   inv(iI - A) = -(A + iI) * inv(A^2 + I)   [A real]
// A(i,j) = -alpha*m_i*m_j*H_f[i,j] + delta_ij * dadd_i   (branch-free rebuild)
//   where dadd_i = alpha*(deg_i + 1e-9 - rmean)*m_i^2 + beta
// Phase 1: M = A*A + I built into LDS with v_wmma_f32_16x16x4_f32
// Phase 2: in-place Gauss-Jordan on LDS (256x257 padded, bank-conflict-free)
// Phase 3: diag: re = -diag(A*M), im = -diag(M)
// ~260 KB dynamic LDS fits CDNA5's 320 KB WGP LDS.
// ---------------------------------------------------------------------------
__global__ void k_invert(const float* __restrict__ H, const float* __restrict__ deg,
                         const int* __restrict__ ptr,
                         const float* __restrict__ evw, const float* __restrict__ evb,
                         const float* __restrict__ evrm, const float* __restrict__ evrv,
                         float* __restrict__ vre, float* __restrict__ vim) {
  extern __shared__ float smem[];
  float* M    = smem;                       // NMAT * LDM
  float* prow = M + NMAT * LDM;             // NMAT
  float* msk  = prow + NMAT;                // NMAT  (mask as float)
  float* dadd = msk + NMAT;                 // NMAT  (diagonal additive)
  int c = blockIdx.x;
  int b = c / FP_, cfp = c % FP_, f = cfp % F_;
  int tid = threadIdx.x;
  float alpha = evw[cfp] * rsqrtf(evrv[cfp] + 1e-5f);   // EPS_EV
  float rmean = evrm[cfp], beta = evb[cfp];
  int lenb = ptr[b + 1] - ptr[b];
  const float* Hf   = H   + ((size_t)(b * F_ + f)) * NMAT * NMAT;
  const float* degf = deg + (size_t)(b * F_ + f) * NMAT;

  // per-channel constants into LDS (once)
  {
    float mi = (tid < lenb) ? 1.0f : 0.0f;
    msk[tid]  = mi;
    dadd[tid] = alpha * (degf[tid] + 1e-9f - rmean) * mi * mi + beta;
  }
  __syncthreads();

  // ---- Phase 1: M = A*A + I via WMMA fp32 (16x16 tiles, K stepped by 4) ----
  int wave = tid >> 5, lane = tid & 31;
  int r = lane & 15;
  int kko = (lane < 16) ? 0 : 2;       // A 16x4 layout: lanes 0-15 K=0,1; 16-31 K=2,3
  for (int tile = wave; tile < 256; tile += 8) {
    int ti = tile >> 4, tj = tile & 15;
    int arow = ti * 16 + r;
    int bcol = tj * 16 + r;
    const float* HrowA = Hf + (size_t)arow * NMAT;
    __builtin_prefetch(HrowA, 0, 1);                     // global_prefetch_b8
    float nu_a = -alpha * msk[arow];                     // hoisted row factors
    float d_a  = dadd[arow];
    float nmb  = -alpha * msk[bcol];
    float d_b  = dadd[bcol];
    v8f acc = {0.f, 0.f, 0.f, 0.f, 0.f, 0.f, 0.f, 0.f};
#pragma unroll 4
    for (int k = 0; k < NMAT; k += 4) {
      int kk = k + kko;
      float m0 = msk[kk], m1 = msk[kk + 1];
      float h0 = HrowA[kk], h1 = HrowA[kk + 1];          // contiguous -> b64
      float g0 = Hf[(size_t)kk * NMAT + bcol];
      float g1 = Hf[(size_t)(kk + 1) * NMAT + bcol];
      v2f af, bf;
      af.x = fmaf(nu_a * m0, h0, (arow == kk)     ? d_a : 0.0f);
      af.y = fmaf(nu_a * m1, h1, (arow == kk + 1) ? d_a : 0.0f);
      bf.x = fmaf(nmb * m0, g0, (kk     == bcol)  ? d_b : 0.0f);
      bf.y = fmaf(nmb * m1, g1, (kk + 1 == bcol)  ? d_b : 0.0f);
      acc = __builtin_amdgcn_wmma_f32_16x16x4_f32(false, af, false, bf, (short)0, acc, false, false);
    }
    int colg  = tj * 16 + (lane & 15);
    int rbase = ti * 16 + ((lane < 16) ? 0 : 8);   // C/D layout: VGPR v -> M=v / M=8+v
    for (int v = 0; v < 8; ++v) {
      int rowg = rbase + v;
      M[rowg * LDM + colg] = acc[v] + ((rowg == colg) ? 1.0f : 0.0f);
    }
  }
  __syncthreads();

  // ---- Phase 2: in-place Gauss-Jordan (no pivoting; A^2+I well-conditioned) ----
  for (int k = 0; k < NMAT; ++k) {
    prow[tid] = M[k * LDM + tid];
    __syncthreads();
    float pinv = 1.0f / prow[k];
    int i = tid;
    if (i == k) {
      for (int j = 0; j < NMAT; ++j) M[i * LDM + j] = prow[j] * pinv;
      M[i * LDM + k] = pinv;
    } else {
      float fct = M[i * LDM + k] * pinv;
      for (int j = 0; j < NMAT; ++j) M[i * LDM + j] -= fct * prow[j];
      M[i * LDM + k] = -fct;
    }
    __syncthreads();
  }

  // ---- Phase 3: diag extraction ----
  int i = tid;
  const float* HrowI = Hf + (size_t)i * NMAT;
  float t = 0.f;
  for (int j = 0; j < NMAT; ++j)   // column read: bank = (j + i) % 64, conflict-free
    t += msk[j] * HrowI[j] * M[j * LDM + i];
  float mii = M[i * LDM + i];
  float s = fmaf(-alpha * msk[i], t, dadd[i] * mii);   // diag(A*M)_i
  vre[((size_t)(b * NG_ + i)) * FP_ + cfp] = -s;
  vim[((size_t)(b * NG_ + i)) * FP_ + cfp] = -mii;
}

// ---------------------------------------------------------------------------
// K5: switch-norm on re/im + final GEMM (2048 x 64 x 128)
// ---------------------------------------------------------------------------
__global__ void k_out(const float* __restrict__ vre, const float* __restrict__ vim,
                      const float* __restrict__ srw, const float* __restrict__ srb,
                      const float* __restrict__ srmw, const float* __restrict__ srvw,
                      const float* __restrict__ srrm, const float* __restrict__ srrv,
                      const float* __restrict__ scw, const float* __restrict__ scb,
                      const float* __restrict__ scmw, const float* __restrict__ scvw,
                      const float* __restrict__ scrm, const float* __restrict__ scrv,
                      const float* __restrict__ Wout, float* __restrict__ out) {
  __shared__ float fre[FP_], fim[FP_], feats[2 * FP_];
  int node = blockIdx.x, t = threadIdx.x;
  if (t < FP_) { fre[t] = vre[(size_t)node * FP_ + t]; fim[t] = vim[(size_t)node * FP_ + t]; }
  __syncthreads();
  float sr = 0.f, sr2 = 0.f, si = 0.f, si2 = 0.f;
  for (int k = 0; k < FP_; ++k) {
    float a = fre[k]; sr += a; sr2 += a * a;
    float q = fim[k]; si += q; si2 += q * q;
  }
  float mr = sr / FP_, mi = si / FP_;
  float vr = (sr2 - FP_ * mr * mr) / (FP_ - 1);   // ddof=1
  float vi = (si2 - FP_ * mi * mi) / (FP_ - 1);
  if (t < FP_) {
    float e0 = __expf(srmw[0]), e1 = __expf(srmw[1]);
    float mw0 = e0 / (e0 + e1), mw1 = e1 / (e0 + e1);
    float f0 = __expf(srvw[0]), f1 = __expf(srvw[1]);
    float vw0 = f0 / (f0 + f1), vw1 = f1 / (f0 + f1);
    float mean = mw0 * mr + mw1 * srrm[t];
    float var  = vw0 * vr + vw1 * srrv[t];
    feats[t] = (fre[t] - mean) * rsqrtf(var + 1e-5f) * srw[t] + srb[t];
    e0 = __expf(scmw[0]); e1 = __expf(scmw[1]); mw0 = e0 / (e0 + e1); mw1 = e1 / (e0 + e1);
    f0 = __expf(scvw[0]); f1 = __expf(scvw[1]); vw0 = f0 / (f0 + f1); vw1 = f1 / (f0 + f1);
    mean = mw0 * mi + mw1 * scrm[t];
    var  = vw0 * vi + vw1 * scrv[t];
    feats[FP_ + t] = (fim[t] - mean) * rsqrtf(var + 1e-5f) * scw[t] + scb[t];
  }
  __syncthreads();
  float acc = 0.f;
  for (int k = 0; k < 2 * FP_; ++k) acc += feats[k] * Wout[k * S_ + t];
  out[(size_t)node * S_ + t] = acc * 0.125f;   // /sqrt(64)
}

// ---------------------------------------------------------------------------
extern "C" void kernel_launch(void* const* d_in, const int* in_sizes, int n_in,
                              void* d_out, int out_size, void* d_ws, size_t ws_size,
                              hipStream_t stream) {
  (void)in_sizes; (void)out_size; (void)ws_size;
  const float* nf   = (const float*)d_in[0];
  const float* ef   = (const float*)d_in[1];
  const int*   eidx = (const int*)d_in[2];
  const int*   bat  = (const int*)d_in[3];
  const int*   ptr  = (const int*)d_in[4];
  const float* Ws   = (const float*)d_in[5];
  const float* bs   = (const float*)d_in[6];
  const float *We1, *We2, *We3, *We4, *Wm1, *Wm2, *Wm3, *evw, *evb, *evrm, *evrv;
  const float *srw, *srb, *srmw, *srvw, *srrm, *srrv;
  const float *scw, *scb, *scmw, *scvw, *scrm, *scrv, *Wout;
  if (n_in >= 31) {   // python lists expanded into individual inputs
    We1 = (const float*)d_in[7];  We2 = (const float*)d_in[8];
    We3 = (const float*)d_in[9];  We4 = (const float*)d_in[10];
    Wm1 = (const float*)d_in[11]; Wm2 = (const float*)d_in[12]; Wm3 = (const float*)d_in[13];
    evw = (const float*)d_in[14]; evb = (const float*)d_in[15];
    evrm = (const float*)d_in[16]; evrv = (const float*)d_in[17];
    srw = (const float*)d_in[18]; srb = (const float*)d_in[19]; srmw = (const float*)d_in[20];
    srvw = (const float*)d_in[21]; srrm = (const float*)d_in[22]; srrv = (const float*)d_in[23];
    scw = (const float*)d_in[24]; scb = (const float*)d_in[25]; scmw = (const float*)d_in[26];
    scvw = (const float*)d_in[27]; scrm = (const float*)d_in[28]; scrv = (const float*)d_in[29];
    Wout = (const float*)d_in[30];
  } else {            // python lists concatenated into one buffer each
    const float* em = (const float*)d_in[7];
    We1 = em; We2 = em + 8 * 64; We3 = We2 + 64 * 64; We4 = We3 + 64 * 64;
    const float* mm = (const float*)d_in[8];
    Wm1 = mm; Wm2 = mm + 128 * 64; Wm3 = Wm2 + 64 * 64;
    evw = (const float*)d_in[9];  evb = (const float*)d_in[10];
    evrm = (const float*)d_in[11]; evrv = (const float*)d_in[12];
    const float* sr = (const float*)d_in[13];
    srw = sr; srb = sr + 32; srmw = sr + 64; srvw = sr + 66; srrm = sr + 68; srrv = sr + 100;
    const float* sc = (const float*)d_in[14];
    scw = sc; scb = sc + 32; scmw = sc + 64; scvw = sc + 66; scrm = sc + 68; scrv = sc + 100;
    Wout = (const float*)d_in[15];
  }

  // workspace layout (~18.4 MB total, L2-resident on MI455X's 192 MB L2)
  float* ws  = (float*)d_ws;
  float* ns  = ws;                                    // 2048*128
  float* H   = ns  + (size_t)N_ * S_;                 // 8*8*256*256
  float* deg = H   + (size_t)B_ * F_ * NMAT * NMAT;   // 16384
  float* vre = deg + (size_t)B_ * F_ * NMAT;          // 65536
  float* vim = vre + (size_t)N_ * FP_;                // 65536

  hipMemsetAsync(H, 0, sizeof(float) * (size_t)B_ * F_ * NMAT * NMAT, stream);

  k_node_s<<<N_, S_, 0, stream>>>(nf, Ws, bs, ns);
  k_edge<<<E_, 64, 0, stream>>>(ef, eidx, bat, ptr, ns, We1, We2, We3, We4, Wm1, Wm2, Wm3, H);
  k_deg<<<B_ * F_ * NMAT, NMAT, 0, stream>>>(H, deg);

  size_t smem = sizeof(float) * (size_t)(NMAT * LDM + 3 * NMAT);   // ~260 KB, fits 320 KB WGP LDS
  hipFuncSetAttribute((const void*)k_invert, hipFuncAttributeMaxDynamicSharedMemorySize, (int)smem);
  k_invert<<<B_ * FP_, 256, smem, stream>>>(H, deg, ptr, evw, evb, evrm, evrv, vre, vim);

  k_out<<<N_, S_, 0, stream>>>(vre, vim, srw, srb, srmw, srvw, srrm, srrv,
                               scw, scb, scmw, scvw, scrm, scrv, Wout, (float*)d_out);
}